// GATHead_51427938402742
// MI455X (gfx1250) — compile-verified
//
#include <hip/hip_runtime.h>
#include <hip/hip_bf16.h>

#define N_NODES 100000
#define N_EDGES 1600000
#define IN_F    128
#define OUT_F   64
#define NEG_SLOPE 0.2f

typedef __attribute__((ext_vector_type(2))) float v2f;
typedef __attribute__((ext_vector_type(8))) float v8f;

// ---------------------------------------------------------------------------
// Init: zero d_out, m = -inf, s = 0   (must run every call; harness poisons)
// ---------------------------------------------------------------------------
__global__ __launch_bounds__(256) void gat_init(float* __restrict__ m,
                                                float* __restrict__ s,
                                                float* __restrict__ out) {
    int i = blockIdx.x * 256 + threadIdx.x;
    if (i < N_NODES * OUT_F) out[i] = 0.0f;
    if (i < N_NODES) {
        m[i] = __int_as_float(0xff800000);  // -inf
        s[i] = 0.0f;
    }
}

// ---------------------------------------------------------------------------
// h = x @ W^T  via V_WMMA_F32_16X16X4_F32 (exact f32 accumulate)
// One wave -> one 16x16 tile of h. 6250 row tiles x 4 col tiles, 8 waves/block.
// f32 16x4 A layout: lane%16 = M; VGPR0/1 hold K = {0,1} (lanes 0-15),
// K = {2,3} (lanes 16-31)  ==> one float2 load per lane per K-step of 4.
// B (= W^T, K x N) mirrors this: lane%16 = N = row of W.
// ---------------------------------------------------------------------------
__global__ __launch_bounds__(256) void gat_gemm(const float* __restrict__ x,
                                                const float* __restrict__ W,
                                                float* __restrict__ h) {
    const int wave = threadIdx.x >> 5;
    const int lane = threadIdx.x & 31;
    const int gw   = blockIdx.x * 8 + wave;   // global wave id
    const int row_tile = gw >> 2;             // 0..6249
    const int col_tile = gw & 3;              // 0..3
    const int half = lane >> 4;               // 0 or 1
    const int lm   = lane & 15;

    const float* xr = x + (size_t)(row_tile * 16 + lm) * IN_F;  // A: node row
    const float* wr = W + (size_t)(col_tile * 16 + lm) * IN_F;  // B: W row = out feature

    v8f acc = {};
#pragma unroll
    for (int k = 0; k < IN_F; k += 4) {
        v2f a2 = *(const v2f*)(xr + k + 2 * half);
        v2f b2 = *(const v2f*)(wr + k + 2 * half);
        acc = __builtin_amdgcn_wmma_f32_16x16x4_f32(
            /*neg_a=*/false, a2, /*neg_b=*/false, b2,
            /*c_mod=*/(short)0, acc, /*reuse_a=*/false, /*reuse_b=*/false);
    }

    // D layout: VGPR r, lanes 0-15 -> (M=r, N=lm); lanes 16-31 -> (M=r+8, N=lm)
    float* hp = h + (size_t)(row_tile * 16) * OUT_F + col_tile * 16 + lm;
#pragma unroll
    for (int r = 0; r < 8; ++r)
        hp[(size_t)(r + 8 * half) * OUT_F] = acc[r];
}

// ---------------------------------------------------------------------------
// Per-node attention logits: asrc[n] = h[n].a[0:64], adst[n] = h[n].a[64:128]
// ---------------------------------------------------------------------------
__global__ __launch_bounds__(256) void gat_alpha(const float* __restrict__ h,
                                                 const float* __restrict__ a,
                                                 float* __restrict__ asrc,
                                                 float* __restrict__ adst) {
    int n = blockIdx.x * 256 + threadIdx.x;
    if (n >= N_NODES) return;
    const float4* hr  = (const float4*)(h + (size_t)n * OUT_F);
    const float4* a0  = (const float4*)(a);
    const float4* a1  = (const float4*)(a + OUT_F);
    float s0 = 0.0f, s1 = 0.0f;
#pragma unroll
    for (int f = 0; f < OUT_F / 4; ++f) {
        float4 hv = hr[f];
        float4 v0 = a0[f];
        float4 v1 = a1[f];
        s0 += hv.x * v0.x + hv.y * v0.y + hv.z * v0.z + hv.w * v0.w;
        s1 += hv.x * v1.x + hv.y * v1.y + hv.z * v1.z + hv.w * v1.w;
    }
    asrc[n] = s0;
    adst[n] = s1;
}

__device__ __forceinline__ float lrelu(float v) {
    return v > 0.0f ? v : NEG_SLOPE * v;
}

// float atomic max via signed-int max / unsigned-int min trick (init = -inf)
__device__ __forceinline__ void atomicMaxF(float* addr, float v) {
    if (v >= 0.0f) atomicMax((int*)addr, __float_as_int(v));
    else           atomicMin((unsigned int*)addr, __float_as_uint(v));
}

// ---------------------------------------------------------------------------
// Edge pass 1: m[src] = segment_max(leaky_relu(asrc[src] + adst[dst]))
// ---------------------------------------------------------------------------
__global__ __launch_bounds__(256) void gat_emax(const int* __restrict__ ei,
                                                const float* __restrict__ asrc,
                                                const float* __restrict__ adst,
                                                float* __restrict__ m) {
    int i = blockIdx.x * 256 + threadIdx.x;
    if (i >= N_EDGES) return;
    int sn = ei[i];
    int dn = ei[N_EDGES + i];
    atomicMaxF(&m[sn], lrelu(asrc[sn] + adst[dn]));
}

// ---------------------------------------------------------------------------
// Edge pass 2: ex = exp(e - m[src]); s[src] += ex (cache ex per edge)
// ---------------------------------------------------------------------------
__global__ __launch_bounds__(256) void gat_esum(const int* __restrict__ ei,
                                                const float* __restrict__ asrc,
                                                const float* __restrict__ adst,
                                                const float* __restrict__ m,
                                                float* __restrict__ s,
                                                float* __restrict__ ex) {
    int i = blockIdx.x * 256 + threadIdx.x;
    if (i >= N_EDGES) return;
    int sn = ei[i];
    int dn = ei[N_EDGES + i];
    float e = lrelu(asrc[sn] + adst[dn]);
    float v = __expf(e - m[sn]);
    ex[i] = v;
    atomicAdd(&s[sn], v);
}

// ---------------------------------------------------------------------------
// Edge pass 3: out[src] += (ex/s[src]) * h[dst]; 64 threads per edge,
// 4 edges per 256-thread block. h (25.6 MB) and out live in 192 MB L2.
// ---------------------------------------------------------------------------
__global__ __launch_bounds__(256) void gat_aggr(const int* __restrict__ ei,
                                                const float* __restrict__ ex,
                                                const float* __restrict__ ssum,
                                                const float* __restrict__ h,
                                                float* __restrict__ out) {
    __shared__ float att_s[4];
    __shared__ int   src_s[4];
    __shared__ int   dst_s[4];
    int t = threadIdx.x;
    int eb = blockIdx.x * 4;
    if (t < 4) {
        int i = eb + t;
        if (i < N_EDGES) {
            int sn = ei[i];
            src_s[t] = sn;
            dst_s[t] = ei[N_EDGES + i];
            att_s[t] = ex[i] / ssum[sn];
        } else {
            src_s[t] = -1;
        }
    }
    __syncthreads();
    int e = t >> 6;
    int f = t & 63;
    int sn = src_s[e];
    if (sn < 0) return;
    float v = att_s[e] * h[(size_t)dst_s[e] * OUT_F + f];
    atomicAdd(&out[(size_t)sn * OUT_F + f], v);
}

// ---------------------------------------------------------------------------
// Launch
// ---------------------------------------------------------------------------
extern "C" void kernel_launch(void* const* d_in, const int* in_sizes, int n_in,
                              void* d_out, int out_size, void* d_ws, size_t ws_size,
                              hipStream_t stream) {
    const float* x  = (const float*)d_in[0];   // (100000,128)
    const int*   ei = (const int*)d_in[1];     // (2,1600000)
    const float* W  = (const float*)d_in[2];   // (64,128)
    const float* a  = (const float*)d_in[3];   // (1,128)
    float* out = (float*)d_out;                // (100000,64)

    // workspace layout (floats): h | asrc | adst | m | s | ex  (~33.6 MB)
    float* ws   = (float*)d_ws;
    float* h    = ws;
    float* asrc = h + (size_t)N_NODES * OUT_F;
    float* adst = asrc + N_NODES;
    float* m    = adst + N_NODES;
    float* s    = m + N_NODES;
    float* exw  = s + N_NODES;

    gat_init <<<(N_NODES * OUT_F + 255) / 256, 256, 0, stream>>>(m, s, out);
    gat_gemm <<<(N_NODES / 16) * (OUT_F / 16) / 8, 256, 0, stream>>>(x, W, h);
    gat_alpha<<<(N_NODES + 255) / 256, 256, 0, stream>>>(h, a, asrc, adst);
    gat_emax <<<(N_EDGES + 255) / 256, 256, 0, stream>>>(ei, asrc, adst, m);
    gat_esum <<<(N_EDGES + 255) / 256, 256, 0, stream>>>(ei, asrc, adst, m, s, exw);
    gat_aggr <<<N_EDGES / 4, 256, 0, stream>>>(ei, exw, s, h, out);
}